// Cookbook_53437983097418
// MI455X (gfx1250) — compile-verified
//
#include <hip/hip_runtime.h>

typedef __attribute__((ext_vector_type(2))) float v2f;
typedef __attribute__((ext_vector_type(8))) float v8f;

#define D        256
#define K        8192
#define NROWS    32768        // 16 * 2048
#define TILE_M   32           // 2 sub-tiles of 16 rows share each B fragment
#define SUBT     (TILE_M / 16)
#define WAVES    8
#define BLOCK    (WAVES * 32)
#define KPW      (K / WAVES)  // 1024 codes per wave
#define LDS_STRIDE 260        // 256 + 4 dword pad -> conflict-free b64 loads
#define BETA     0.1f

// ---------------------------------------------------------------------------
// Main kernel: one block per 32-row z tile. Distance matmul via f32 WMMA with
// 2-way M blocking (one B fragment feeds two accumulators), per-wave shuffle
// argmin reduction, codebook gather (straight-through forward value) and a
// per-block partial of sum((z_q - z)^2).
// ---------------------------------------------------------------------------
__global__ __launch_bounds__(BLOCK)
void vq_argmin_kernel(const float* __restrict__ z,
                      const float* __restrict__ w,
                      float* __restrict__ out,
                      float* __restrict__ partial) {
    __shared__ float zt[TILE_M * LDS_STRIDE];     // ~33 KB z tile
    __shared__ float redV[WAVES * TILE_M];        // 1 KB per-wave row minima
    __shared__ int   redI[WAVES * TILE_M];        // 1 KB per-wave row indices
    __shared__ int   bestIdx[TILE_M];
    __shared__ float lsum[BLOCK];

    const int tid  = threadIdx.x;
    const int lane = tid & 31;
    const int wave = tid >> 5;
    const int row0 = blockIdx.x * TILE_M;

    // Stage the 32x256 z tile into padded LDS (uniform loop -> EXEC stays full).
    for (int i = tid; i < TILE_M * D; i += BLOCK) {
        int r = i >> 8;
        int c = i & 255;
        zt[r * LDS_STRIDE + c] = z[(size_t)(row0 + r) * D + c];
    }
    __syncthreads();

    // WMMA fragment coordinates (ISA 7.12.2, 32-bit 16x4 A / 4x16 B):
    //   lanes 0-15 hold K sub-pair {0,1}, lanes 16-31 hold {2,3};
    //   v2f element 0 = first K of pair, element 1 = second.
    const int m    = lane & 15;          // A row / B column within tile
    const int half = lane >> 4;
    const int koff = half * 2;           // K sub-offset for this half-wave

    float minv[SUBT][8];
    int   mini[SUBT][8];
#pragma unroll
    for (int s = 0; s < SUBT; ++s)
#pragma unroll
        for (int v = 0; v < 8; ++v) { minv[s][v] = 3.4e38f; mini[s][v] = 0; }

    // Each wave scans its private 1024-code range: 64 tiles of 16 codes.
    for (int t = 0; t < KPW / 16; ++t) {
        const int n0 = wave * KPW + t * 16;
        const int n  = n0 + m;                       // this lane's code id
        const float* wr = w + (size_t)n * D;

        v8f acc0 = {};
        v8f acc1 = {};
        float wn = 0.f;
#pragma unroll 4
        for (int kc = 0; kc < D; kc += 4) {
            v2f b;                                   // shared B fragment
            b.x = wr[kc + koff];
            b.y = wr[kc + koff + 1];
            wn += b.x * b.x + b.y * b.y;             // ||w_n||^2 (half per half-wave)
            v2f a0, a1;
            a0.x = zt[m * LDS_STRIDE + kc + koff];
            a0.y = zt[m * LDS_STRIDE + kc + koff + 1];
            a1.x = zt[(16 + m) * LDS_STRIDE + kc + koff];
            a1.y = zt[(16 + m) * LDS_STRIDE + kc + koff + 1];
            acc0 = __builtin_amdgcn_wmma_f32_16x16x4_f32(
                false, a0, false, b, (short)0, acc0, false, false);
            acc1 = __builtin_amdgcn_wmma_f32_16x16x4_f32(
                false, a1, false, b, (short)0, acc1, false, false);
        }
        // Complete ||w_n||^2: other half-wave holds the complementary K's.
        wn += __shfl_xor(wn, 16, 32);

        // C layout: lane -> N, VGPR v -> M = v + 8*(lane>>4). Same n per lane.
#pragma unroll
        for (int v = 0; v < 8; ++v) {
            float d0 = wn - 2.0f * acc0[v];
            if (d0 < minv[0][v]) { minv[0][v] = d0; mini[0][v] = n; }
            float d1 = wn - 2.0f * acc1[v];
            if (d1 < minv[1][v]) { minv[1][v] = d1; mini[1][v] = n; }
        }
    }

    // In-wave argmin: butterfly over the 16 lanes of each half (candidates for
    // row M = v + 8*half all live in one half). Index tie-break => first-match
    // semantics like jnp.argmin.
#pragma unroll
    for (int s = 0; s < SUBT; ++s) {
#pragma unroll
        for (int v = 0; v < 8; ++v) {
            float val = minv[s][v];
            int   idx = mini[s][v];
#pragma unroll
            for (int off = 1; off < 16; off <<= 1) {
                float ov = __shfl_xor(val, off, 32);
                int   oi = __shfl_xor(idx, off, 32);
                if (ov < val || (ov == val && oi < idx)) { val = ov; idx = oi; }
            }
            if ((lane & 15) == 0) {                  // one writer per half
                int row = s * 16 + v + 8 * half;     // row within tile
                redV[wave * TILE_M + row] = val;
                redI[wave * TILE_M + row] = idx;
            }
        }
    }
    __syncthreads();

    // Cross-wave cleanup: 8 candidates per row.
    if (tid < TILE_M) {
        float bv = 3.4e38f;
        int   bi = 0x7FFFFFFF;
        for (int wv = 0; wv < WAVES; ++wv) {
            float val = redV[wv * TILE_M + tid];
            int   id  = redI[wv * TILE_M + tid];
            if (val < bv || (val == bv && id < bi)) { bv = val; bi = id; }
        }
        bestIdx[tid] = bi;
    }
    __syncthreads();

    // Gather + straight-through output + local loss partial.
    float ls = 0.f;
    for (int i = tid; i < TILE_M * D; i += BLOCK) {
        int r = i >> 8;
        int c = i & 255;
        float q = w[(size_t)bestIdx[r] * D + c];
        out[(size_t)(row0 + r) * D + c] = q;
        float diff = q - zt[r * LDS_STRIDE + c];
        ls += diff * diff;
    }
    lsum[tid] = ls;
    __syncthreads();
    for (int s = BLOCK / 2; s > 0; s >>= 1) {
        if (tid < s) lsum[tid] += lsum[tid + s];
        __syncthreads();
    }
    if (tid == 0) partial[blockIdx.x] = lsum[0];
}

// ---------------------------------------------------------------------------
// Deterministic final reduction of per-block partials -> scalar loss.
// ---------------------------------------------------------------------------
__global__ __launch_bounds__(256)
void vq_loss_finalize(const float* __restrict__ partial,
                      float* __restrict__ out_loss, int nparts) {
    __shared__ float s[256];
    float acc = 0.f;
    for (int i = threadIdx.x; i < nparts; i += 256) acc += partial[i];
    s[threadIdx.x] = acc;
    __syncthreads();
    for (int st = 128; st > 0; st >>= 1) {
        if (threadIdx.x < st) s[threadIdx.x] += s[threadIdx.x + st];
        __syncthreads();
    }
    if (threadIdx.x == 0)
        *out_loss = (1.0f + BETA) * s[0] / (float)(NROWS * D);
}

extern "C" void kernel_launch(void* const* d_in, const int* in_sizes, int n_in,
                              void* d_out, int out_size, void* d_ws, size_t ws_size,
                              hipStream_t stream) {
    (void)in_sizes; (void)n_in; (void)out_size; (void)ws_size;
    const float* z = (const float*)d_in[0];   // [16*2048*256] f32
    const float* w = (const float*)d_in[1];   // [8192*256]    f32
    float* out     = (float*)d_out;           // [N*D] z_q_st, then 1 loss scalar
    float* partial = (float*)d_ws;            // 1024 f32 partials (4 KB)

    const int nblocks = NROWS / TILE_M;       // 1024
    vq_argmin_kernel<<<nblocks, BLOCK, 0, stream>>>(z, w, out, partial);
    vq_loss_finalize<<<1, 256, 0, stream>>>(partial, out + (size_t)NROWS * D, nblocks);
}